// EntropyModel_4655744549502
// MI455X (gfx1250) — compile-verified
//
#include <hip/hip_runtime.h>
#include <cstdint>

// Problem constants (from reference)
constexpr int B_  = 2;
constexpr int C_  = 128;
constexpr int Hh  = 192;
constexpr int Ww  = 192;
constexpr int K_  = 4;
constexpr int O_  = 60;
constexpr int HW  = Hh * Ww;           // 36864
constexpr int TILE = 32;               // pixels per wave (2 WMMA n-tiles)
constexpr int NT   = 2;                // n-tiles per wave
constexpr int MAX_TILES = HW / TILE;   // 1152 (worst case one bucket owns all pixels)
constexpr int LSTR = 132;              // padded LDS row stride (floats) for bank spread

typedef float v8f __attribute__((ext_vector_type(8)));
typedef float v2f __attribute__((ext_vector_type(2)));

// ---------------------------------------------------------------------------
// Bucketing: counts[b*K+k], idx[(b*K+k)*HW + slot] = pixel
// ---------------------------------------------------------------------------
__global__ void zero_counts_kernel(int* counts) {
    if (threadIdx.x < B_ * K_) counts[threadIdx.x] = 0;
}

__global__ void scatter_seg_kernel(const long long* __restrict__ seg,
                                   int* __restrict__ counts,
                                   int* __restrict__ idx) {
    int t = blockIdx.x * blockDim.x + threadIdx.x;
    if (t >= B_ * HW) return;
    int b = t / HW;
    int p = t - b * HW;
    int k = (int)seg[t];                 // seg is int64 in the reference
    int bk = b * K_ + k;
    int slot = atomicAdd(&counts[bk], 1);
    idx[bk * HW + slot] = p;
}

// ---------------------------------------------------------------------------
// WMMA f32 16x16x4 wrapper
// A (16x4):  lane l holds row (l&15); regs = K k0+2*(l>>4), k0+2*(l>>4)+1
// B (4x16):  lane l holds col (l&15); same K split across halves
// C/D:       reg r of lane l = element (M = r + 8*(l>>4), N = l&15)
// ---------------------------------------------------------------------------
__device__ __forceinline__ v8f wmma4(v2f a, v2f b, v8f c) {
    return __builtin_amdgcn_wmma_f32_16x16x4_f32(
        /*neg_a=*/false, a, /*neg_b=*/false, b,
        /*c_mod=*/(short)0, c, /*reuse_a=*/false, /*reuse_b=*/false);
}

// One GEMM stage over a 32-pixel tile:
//   outT[n][m] = act( Wm[m][:] . inT[n][:] + bias[m] ) (+ outT_prev if RESID)
// inT/outT are LDS tiles [32][LSTR]; MT m-tiles of 16 output channels each.
// Each A fragment (weights, from global/L1) is reused for NT=2 WMMAs.
template <int MT, bool LRELU, bool RESID>
__device__ __forceinline__ void stage_lds(const float* __restrict__ Wm,
                                          const float* __restrict__ bias,
                                          const float* inT, float* outT,
                                          int lm, int half) {
    v8f acc[NT][MT];
#pragma unroll
    for (int mt = 0; mt < MT; ++mt) {
        const float* bp = bias + mt * 16 + 8 * half;
#pragma unroll
        for (int r = 0; r < 8; ++r) {
            float bv = bp[r];
            acc[0][mt][r] = bv;
            acc[1][mt][r] = bv;
        }
    }

#pragma unroll 2
    for (int k0 = 0; k0 < C_; k0 += 4) {
        v2f bfr[NT];
#pragma unroll
        for (int nt = 0; nt < NT; ++nt) {
            const float* q = inT + (nt * 16 + lm) * LSTR + k0 + 2 * half;
            bfr[nt][0] = q[0];
            bfr[nt][1] = q[1];
        }
#pragma unroll
        for (int mt = 0; mt < MT; ++mt) {
            const float* ap = Wm + (size_t)(mt * 16 + lm) * C_ + k0 + 2 * half;
            v2f afr;
            afr[0] = ap[0];
            afr[1] = ap[1];
            acc[0][mt] = wmma4(afr, bfr[0], acc[0][mt]);
            acc[1][mt] = wmma4(afr, bfr[1], acc[1][mt]);
        }
    }

    __syncthreads();  // single-wave WG: lowers to S_NOP; enforces LDS ordering
#pragma unroll
    for (int nt = 0; nt < NT; ++nt) {
#pragma unroll
        for (int mt = 0; mt < MT; ++mt) {
            float* o = outT + (nt * 16 + lm) * LSTR + mt * 16 + 8 * half;
#pragma unroll
            for (int r = 0; r < 8; ++r) {
                float v = acc[nt][mt][r];
                if (LRELU) v = (v >= 0.0f) ? v : 0.01f * v;
                if (RESID) v += o[r];  // residual: old contents of outT (H1)
                o[r] = v;
            }
        }
    }
    __syncthreads();
}

// ---------------------------------------------------------------------------
// Main kernel: one wave32 per 32-pixel tile of one (b,k) bucket.
// Chain: H1=W1x+b1; U=lrelu(Wr1 H1+br1); H2=H1+Wr2 U+br2; V=lrelu(W3 H2+b3);
//        out = W4 V + b4 (60 rows), scattered back per pixel.
// ---------------------------------------------------------------------------
__global__ __launch_bounds__(32) void egemm_kernel(
    const float* __restrict__ x, const int* __restrict__ counts,
    const int* __restrict__ idx,
    const float* __restrict__ W1, const float* __restrict__ b1,
    const float* __restrict__ Wr1, const float* __restrict__ br1,
    const float* __restrict__ Wr2, const float* __restrict__ br2,
    const float* __restrict__ W3, const float* __restrict__ b3,
    const float* __restrict__ W4, const float* __restrict__ b4,
    float* __restrict__ out) {
    __shared__ float Xb[TILE * LSTR];
    __shared__ float Hb[TILE * LSTR];
    __shared__ int pixs[TILE];

    int bk = blockIdx.x / MAX_TILES;
    int tile = blockIdx.x - bk * MAX_TILES;
    int cnt = counts[bk];
    int n0 = tile * TILE;
    if (n0 >= cnt) return;  // empty tile: whole wave exits (EXEC all-ones for WMMA)

    int b = bk / K_;
    int k = bk - b * K_;
    int lane = threadIdx.x;
    int lm = lane & 15;
    int half = lane >> 4;

    {
        int j = min(n0 + lane, cnt - 1);  // tail: duplicate last pixel (identical writes)
        pixs[lane] = idx[bk * HW + j];
    }
    __syncthreads();

    // Stage gathered input columns into LDS as xT[n][c].
    // All 32 lanes touch the SAME channel plane per iteration -> bucketed pixels
    // are clustered within the plane, so cachelines are shared across lanes.
    {
        int p = pixs[lane];
        const float* xb = x + (size_t)b * C_ * HW + p;
        float* xrow = Xb + lane * LSTR;
#pragma unroll 8
        for (int c = 0; c < C_; ++c)
            xrow[c] = xb[(size_t)c * HW];
    }
    __syncthreads();

    const float* W1k  = W1  + (size_t)k * C_ * C_;
    const float* Wr1k = Wr1 + (size_t)k * C_ * C_;
    const float* Wr2k = Wr2 + (size_t)k * C_ * C_;
    const float* W3k  = W3  + (size_t)k * C_ * C_;

    stage_lds<8, false, false>(W1k,  b1  + k * C_, Xb, Hb, lm, half);  // H1 -> Hb
    stage_lds<8, true,  false>(Wr1k, br1 + k * C_, Hb, Xb, lm, half);  // U  -> Xb
    stage_lds<8, false, true >(Wr2k, br2 + k * C_, Xb, Hb, lm, half);  // H2 = H1 + R -> Hb
    stage_lds<8, true,  false>(W3k,  b3  + k * C_, Hb, Xb, lm, half);  // V  -> Xb

    // Final stage: W4 (60x128) -> global scatter. 4 m-tiles cover m=0..63; rows >=60 clamped.
    {
        v8f acc[NT][4];
#pragma unroll
        for (int mt = 0; mt < 4; ++mt) {
#pragma unroll
            for (int r = 0; r < 8; ++r) {
                int m = mt * 16 + 8 * half + r;
                float bv = b4[k * O_ + min(m, O_ - 1)];
                acc[0][mt][r] = bv;
                acc[1][mt][r] = bv;
            }
        }
#pragma unroll 2
        for (int k0 = 0; k0 < C_; k0 += 4) {
            v2f bfr[NT];
#pragma unroll
            for (int nt = 0; nt < NT; ++nt) {
                const float* q = Xb + (nt * 16 + lm) * LSTR + k0 + 2 * half;
                bfr[nt][0] = q[0];
                bfr[nt][1] = q[1];
            }
#pragma unroll
            for (int mt = 0; mt < 4; ++mt) {
                int row = min(mt * 16 + lm, O_ - 1);  // in-bounds reads; extra rows discarded
                const float* ap = W4 + (size_t)k * O_ * C_ + (size_t)row * C_ + k0 + 2 * half;
                v2f afr;
                afr[0] = ap[0];
                afr[1] = ap[1];
                acc[0][mt] = wmma4(afr, bfr[0], acc[0][mt]);
                acc[1][mt] = wmma4(afr, bfr[1], acc[1][mt]);
            }
        }

        int nvalid = cnt - n0;
        const float* obase = out + (size_t)b * O_ * HW;
#pragma unroll
        for (int nt = 0; nt < NT; ++nt) {
            int n = nt * 16 + lm;
            int p = pixs[n];
            bool live = (n < nvalid);
            float* ob = (float*)obase + p;
            if (live) {
#pragma unroll
                for (int mt = 0; mt < 4; ++mt) {
#pragma unroll
                    for (int r = 0; r < 8; ++r) {
                        int m = mt * 16 + 8 * half + r;
                        if (m < O_) ob[(size_t)m * HW] = acc[nt][mt][r];
                    }
                }
            }
        }
    }
}

// ---------------------------------------------------------------------------
extern "C" void kernel_launch(void* const* d_in, const int* in_sizes, int n_in,
                              void* d_out, int out_size, void* d_ws, size_t ws_size,
                              hipStream_t stream) {
    const float*     x   = (const float*)d_in[0];
    const long long* seg = (const long long*)d_in[1];
    const float* W1  = (const float*)d_in[2];
    const float* b1  = (const float*)d_in[3];
    const float* Wr1 = (const float*)d_in[4];
    const float* br1 = (const float*)d_in[5];
    const float* Wr2 = (const float*)d_in[6];
    const float* br2 = (const float*)d_in[7];
    const float* W3  = (const float*)d_in[8];
    const float* b3  = (const float*)d_in[9];
    const float* W4  = (const float*)d_in[10];
    const float* b4  = (const float*)d_in[11];
    float* out = (float*)d_out;

    // Workspace: counts (8 ints, padded to 64) + idx (B*K*HW ints ~= 1.18 MB)
    int* counts = (int*)d_ws;
    int* idx    = (int*)d_ws + 64;

    zero_counts_kernel<<<1, 32, 0, stream>>>(counts);
    scatter_seg_kernel<<<(B_ * HW + 255) / 256, 256, 0, stream>>>(seg, counts, idx);
    egemm_kernel<<<B_ * K_ * MAX_TILES, 32, 0, stream>>>(
        x, counts, idx, W1, b1, Wr1, br1, Wr2, br2, W3, b3, W4, b4, out);
}